// RnCLoss_18270790877689
// MI455X (gfx1250) — compile-verified
//
#include <hip/hip_runtime.h>
#include <hip/hip_bf16.h>

// ---------------------------------------------------------------------------
// RnC loss for MI455X (gfx1250).  n = 2048 (two views stacked), d = 256.
//   Phase A: G = X X^T strip via V_WMMA_F32_16X16X4_F32 (f32 matrix pipe),
//            e = exp(-sqrt(max(d2,0))/T) staged in LDS.
//   Phase B: denominators via global label-argsort + per-row prefix sums +
//            two binary searches per k (exact same float predicate as the
//            reference's sort/searchsorted, so tie semantics match).
// ---------------------------------------------------------------------------

typedef __attribute__((ext_vector_type(2))) float v2f;
typedef __attribute__((ext_vector_type(8))) float v8f;

#define RNC_N   2048   // 2 * batch
#define RNC_B   1024   // batch
#define RNC_D   256    // feature dim

// Row r of the logically-packed [2048,256] matrix inside features[1024,2,256].
__device__ __forceinline__ const float* xrow(const float* __restrict__ f, int r) {
    return f + ((r < RNC_B) ? (r * (2 * RNC_D)) : ((r - RNC_B) * (2 * RNC_D) + RNC_D));
}

// ---------------------------------------------------------------------------
// Kernel 1: per-row squared norms, tiled labels, zero the accumulators.
// One wave per row; 8 rows per 256-thread block; 256 blocks.
// ---------------------------------------------------------------------------
__global__ __launch_bounds__(256) void rnc_prep(const float* __restrict__ feat,
                                                const float* __restrict__ labels,
                                                float* __restrict__ sq,
                                                float* __restrict__ lab,
                                                float* __restrict__ acc) {
    if (blockIdx.x == 0 && threadIdx.x == 0) { acc[0] = 0.0f; acc[1] = 0.0f; }
    const int row  = blockIdx.x * 8 + (threadIdx.x >> 5);
    const int lane = threadIdx.x & 31;
    const float* p = xrow(feat, row);
    float s = 0.0f;
#pragma unroll
    for (int t = 0; t < RNC_D / 32; ++t) {
        float x = p[lane + 32 * t];
        s += x * x;
    }
#pragma unroll
    for (int off = 16; off; off >>= 1) s += __shfl_xor(s, off, 32);
    if (lane == 0) {
        sq[row]  = s;
        lab[row] = labels[row & (RNC_B - 1)];
    }
}

// ---------------------------------------------------------------------------
// Kernel 2: global bitonic argsort of the 2048 labels (one workgroup).
// slab[] = sorted labels, perm[] = argsort permutation.
// ---------------------------------------------------------------------------
__global__ __launch_bounds__(1024) void rnc_sortlab(const float* __restrict__ lab,
                                                    float* __restrict__ slab,
                                                    int* __restrict__ perm) {
    __shared__ float k_[RNC_N];
    __shared__ int   v_[RNC_N];
    const int tid = threadIdx.x;
    for (int t = tid; t < RNC_N; t += 1024) { k_[t] = lab[t]; v_[t] = t; }
    __syncthreads();
    for (int size = 2; size <= RNC_N; size <<= 1) {
        for (int stride = size >> 1; stride > 0; stride >>= 1) {
            const int lo  = 2 * tid - (tid & (stride - 1));
            const int hi  = lo + stride;
            const bool asc = (lo & size) == 0;
            float ka = k_[lo], kb = k_[hi];
            if (asc ? (ka > kb) : (ka < kb)) {
                k_[lo] = kb; k_[hi] = ka;
                int va = v_[lo]; v_[lo] = v_[hi]; v_[hi] = va;
            }
            __syncthreads();
        }
    }
    for (int t = tid; t < RNC_N; t += 1024) { slab[t] = k_[t]; perm[t] = v_[t]; }
}

// ---------------------------------------------------------------------------
// Kernel 3: one workgroup per 16-row anchor strip (128 workgroups, 256 thr).
// ---------------------------------------------------------------------------
__global__ __launch_bounds__(256) void rnc_strip(const float* __restrict__ feat,
                                                 const float* __restrict__ sq,
                                                 const float* __restrict__ lab,
                                                 const float* __restrict__ slab,
                                                 const int* __restrict__ perm,
                                                 float* __restrict__ acc) {
    __shared__ float eLds[16 * RNC_N];    // 128 KB: exp(logits) for the strip
    __shared__ float slabLds[RNC_N];      //   8 KB: label values, sorted
    __shared__ float csum[RNC_N + 1];     //   8 KB: per-row prefix sums
    __shared__ float aux[256];
    __shared__ float sqA[16];
    __shared__ float red[16];

    const int tid  = threadIdx.x;
    const int lane = tid & 31;
    const int wave = tid >> 5;
    const int i0   = blockIdx.x * 16;     // first anchor row of this strip

    for (int t = tid; t < RNC_N; t += 256) slabLds[t] = slab[t];
    if (tid < 16) sqA[tid] = sq[i0 + tid];
    __syncthreads();

    // ================= Phase A: WMMA f32 GEMM strip =================
    // Lane mapping (ISA 7.12.2, 32-bit A 16x4 / B 4x16).
    const int mr = lane & 15;             // M (A) / N (B) index for this lane
    const int kh = (lane >> 4) * 2;       // K sub-offset: lanes 16-31 hold K+2

    const float* arow = xrow(feat, i0 + mr);
    v2f a[RNC_D / 4];                     // anchor tile: 64 v2f in registers
#pragma unroll
    for (int kk = 0; kk < RNC_D / 4; ++kk) {
        const float* ap = arow + kk * 4 + kh;
        a[kk] = v2f{ap[0], ap[1]};
    }

    float logitSum = 0.0f;

    for (int ct = wave; ct < RNC_N / 16; ct += 8) {
        const int    jcol = ct * 16 + mr;
        const float* brow = xrow(feat, jcol);   // B[k][n] = X[jcol][k]
        v8f cacc = {};
#pragma unroll
        for (int kk = 0; kk < RNC_D / 4; ++kk) {
            const float* bp = brow + kk * 4 + kh;
            v2f b = v2f{bp[0], bp[1]};
            cacc = __builtin_amdgcn_wmma_f32_16x16x4_f32(
                false, a[kk], false, b, (short)0, cacc, false, false);
        }
        const float sqj = sq[jcol];
#pragma unroll
        for (int v = 0; v < 8; ++v) {
            const int m = v + (lane >> 4) * 8;  // C/D layout: VGPR v, lane half
            const int i = i0 + m;
            float d2 = fmaxf(sqA[m] + sqj - 2.0f * cacc[v], 0.0f);
            float lg = -0.5f * __builtin_sqrtf(d2);   // / TEMPERATURE(=2)
            float e  = __expf(lg);
            if (jcol == i) e = 0.0f;                  // drop diagonal
            else           logitSum += lg;
            eLds[m * RNC_N + jcol] = e;
        }
    }
    __syncthreads();

    // ================= Phase B: denominators via sorted windows =============
    // denom_k = total - sum_{j : |lab_i-lab_j| < ld_k} e_j.  Along the label-
    // sorted axis, ld_j is V-shaped around pos = lower_bound(slab, lab_i), so
    // the strict-inside set is a contiguous window [L, R) found by two binary
    // searches evaluating the exact reference predicate.
    float logDenomSum = 0.0f;

    for (int r = 0; r < 16; ++r) {
        const int   i    = i0 + r;
        const float labi = lab[i];
        const float* eRow = &eLds[r * RNC_N];

        // gather e into sorted order + local inclusive prefix (8 per thread)
        float loc[8];
        {
            const int base = tid * 8;
            const int4* pp = (const int4*)(perm + base);
            int4 p0 = pp[0], p1 = pp[1];
            float run = 0.0f;
            run += eRow[p0.x]; loc[0] = run;
            run += eRow[p0.y]; loc[1] = run;
            run += eRow[p0.z]; loc[2] = run;
            run += eRow[p0.w]; loc[3] = run;
            run += eRow[p1.x]; loc[4] = run;
            run += eRow[p1.y]; loc[5] = run;
            run += eRow[p1.z]; loc[6] = run;
            run += eRow[p1.w]; loc[7] = run;
            aux[tid] = run;
        }
        __syncthreads();
        // block scan of aux (wave 0, 8 chunks of 32 with shfl scans)
        if (wave == 0) {
            float off = 0.0f;
            for (int c = 0; c < 8; ++c) {
                float x = aux[c * 32 + lane];
#pragma unroll
                for (int d = 1; d < 32; d <<= 1) {
                    float y = __shfl_up(x, d, 32);
                    if (lane >= d) x += y;
                }
                float tot = __shfl(x, 31, 32);
                aux[c * 32 + lane] = off + x;   // inclusive across block
                off += tot;
            }
        }
        __syncthreads();
        const float texcl = (tid == 0) ? 0.0f : aux[tid - 1];
        if (tid == 0) csum[0] = 0.0f;
#pragma unroll
        for (int q = 0; q < 8; ++q) csum[tid * 8 + q + 1] = texcl + loc[q];
        __syncthreads();

        const float total = csum[RNC_N];

        // pivot: first index with slab[j] >= labi (uniform across block)
        int plo = 0, phi = RNC_N;
        while (plo < phi) {
            int mid = (plo + phi) >> 1;
            if (slabLds[mid] < labi) plo = mid + 1; else phi = mid;
        }
        const int pos = plo;

#pragma unroll
        for (int kt = 0; kt < 8; ++kt) {
            const int   k   = tid + 256 * kt;
            const float ldk = fabsf(labi - lab[k]);
            // Left: first L in [0,pos) with (labi - slab[L]) < ldk
            int a0 = 0, b0 = pos;
            while (a0 < b0) {
                int mid = (a0 + b0) >> 1;
                if ((labi - slabLds[mid]) < ldk) b0 = mid; else a0 = mid + 1;
            }
            const int Lb = a0;
            // Right: first R in [pos,N) with (slab[R] - labi) >= ldk
            a0 = pos; b0 = RNC_N;
            while (a0 < b0) {
                int mid = (a0 + b0) >> 1;
                if ((slabLds[mid] - labi) < ldk) a0 = mid + 1; else b0 = mid;
            }
            const int Rb = a0;
            const float denom = total - (csum[Rb] - csum[Lb]);
            if (k != i) logDenomSum += __logf(denom);
        }
        __syncthreads();   // csum/aux reused next row
    }

    // ---- block reduction + single atomic per accumulator ----
#pragma unroll
    for (int off = 16; off; off >>= 1) {
        logitSum    += __shfl_xor(logitSum, off, 32);
        logDenomSum += __shfl_xor(logDenomSum, off, 32);
    }
    if (lane == 0) { red[wave] = logDenomSum; red[8 + wave] = logitSum; }
    __syncthreads();
    if (tid == 0) {
        float sd = 0.0f, sl = 0.0f;
#pragma unroll
        for (int w = 0; w < 8; ++w) { sd += red[w]; sl += red[8 + w]; }
        atomicAdd(&acc[0], sd);
        atomicAdd(&acc[1], sl);
    }
}

// ---------------------------------------------------------------------------
// Kernel 4: loss = (sum log denom - sum logits) / (n*(n-1))
// ---------------------------------------------------------------------------
__global__ void rnc_final(const float* __restrict__ acc, float* __restrict__ out) {
    if (threadIdx.x == 0 && blockIdx.x == 0) {
        out[0] = (acc[0] - acc[1]) *
                 (1.0f / ((float)RNC_N * (float)(RNC_N - 1)));
    }
}

// ---------------------------------------------------------------------------
extern "C" void kernel_launch(void* const* d_in, const int* in_sizes, int n_in,
                              void* d_out, int out_size, void* d_ws, size_t ws_size,
                              hipStream_t stream) {
    (void)in_sizes; (void)n_in; (void)out_size; (void)ws_size;
    const float* feat   = (const float*)d_in[0];   // [1024, 2, 256] f32
    const float* labels = (const float*)d_in[1];   // [1024, 1]      f32
    // d_in[2] (ranks, int64) is unused by the reference forward.

    float* ws   = (float*)d_ws;
    float* sq   = ws;                    // 2048 floats
    float* lab  = ws + RNC_N;            // 2048 floats
    float* acc  = ws + 2 * RNC_N;        // acc[0]=sum log denom, acc[1]=sum lg
    float* slab = ws + 2 * RNC_N + 8;    // 2048 floats (32B aligned)
    int*   perm = (int*)(ws + 3 * RNC_N + 8);  // 2048 ints (32B aligned)

    rnc_prep   <<<RNC_N / 8, 256, 0, stream>>>(feat, labels, sq, lab, acc);
    rnc_sortlab<<<1, 1024, 0, stream>>>(lab, slab, perm);
    rnc_strip  <<<RNC_N / 16, 256, 0, stream>>>(feat, sq, lab, slab, perm, acc);
    rnc_final  <<<1, 64, 0, stream>>>(acc, (float*)d_out);
}